// TopKSoftmaxGate_tensorflow_69037304316405
// MI455X (gfx1250) — compile-verified
//
#include <hip/hip_runtime.h>
#include <hip/hip_bf16.h>

typedef __attribute__((ext_vector_type(2))) float v2f;
typedef __attribute__((ext_vector_type(8))) float v8f;

#define B_TOTAL   8192
#define D_DIM     768
#define E_DIM     16
#define P_DIM     4
#define TILE_ROWS 16

// Fused MoE gate + combine.
// Block = 256 threads (8 waves), each block owns TILE_ROWS rows of B.
// Wave 0: router logits via V_WMMA_F32_16X16X4_F32.
// All waves: stream h tile (768 KB/block) once, fully coalesced.
__global__ __launch_bounds__(256)
void moe_fused_kernel(const float* __restrict__ h,
                      const float* __restrict__ x,
                      const float* __restrict__ ew,     // [E, D]
                      const float* __restrict__ bias,   // [E]
                      const float* __restrict__ perm,   // [P, E, E]
                      float* __restrict__ y,            // [B, D]
                      float* __restrict__ accum)        // [32] soft/hard sums
{
    __shared__ float lds_pm[E_DIM][E_DIM];              // P_mean
    __shared__ float lds_logits[TILE_ROWS][E_DIM + 1];
    __shared__ float lds_p1[TILE_ROWS], lds_p2[TILE_ROWS];
    __shared__ int   lds_i1[TILE_ROWS], lds_i2[TILE_ROWS];
    __shared__ float lds_sperm[TILE_ROWS][E_DIM + 1];

    const int tid = threadIdx.x;
    const int r0  = blockIdx.x * TILE_ROWS;

    // ---- P_mean: one entry per thread (256 = 16x16) ----
    {
        const int e = tid >> 4, j = tid & 15;
        float s = 0.0f;
#pragma unroll
        for (int p = 0; p < P_DIM; ++p)
            s += perm[(p * E_DIM + e) * E_DIM + j];
        lds_pm[e][j] = s * 0.25f;
    }

    // ---- Phase 1: router logits (wave 0 only) ----
    if (tid < 32) {
        const int lane = tid;
        const int half = lane >> 4;     // 0: K={0,1}, 1: K={2,3}
        const int lr   = lane & 15;     // M (for A) / N (for B)

        const float* xrow = x  + (size_t)(r0 + lr) * D_DIM + half * 2;
        const float* wrow = ew + (size_t)lr        * D_DIM + half * 2;

#if __has_builtin(__builtin_amdgcn_wmma_f32_16x16x4_f32)
        v8f c = {};
        for (int k = 0; k < D_DIM; k += 4) {
            v2f a, b;
            a[0] = xrow[k]; a[1] = xrow[k + 1];     // A: 16x4 fragment
            b[0] = wrow[k]; b[1] = wrow[k + 1];     // B: 4x16 fragment (= W.T)
            c = __builtin_amdgcn_wmma_f32_16x16x4_f32(
                    false, a, false, b, (short)0, c, false, false);
        }
        const float bn = bias[lr];
#pragma unroll
        for (int v = 0; v < 8; ++v) {
            const int row = v + 8 * half;           // C/D layout: VGPR v, half
            lds_logits[row][lr] = c[v] + bn;        // element (M=row, N=lr)
        }
#else
        // Fallback (no f32 WMMA builtin): scalar dot per (row, expert).
        // lane < 16 handles row=lane; lanes 16-31 idle.
        if (half == 0) {
            for (int e = 0; e < E_DIM; ++e) {
                const float* xr = x  + (size_t)(r0 + lr) * D_DIM;
                const float* wr = ew + (size_t)e * D_DIM;
                float acc = bias[e];
                for (int k = 0; k < D_DIM; ++k) acc += xr[k] * wr[k];
                lds_logits[lr][e] = acc;
            }
        }
#endif
    }
    __syncthreads();

    // ---- Phase 2: top-2 + softmax per row (16 threads) ----
    if (tid < TILE_ROWS) {
        const int r = tid;
        float v1 = -__builtin_huge_valf(), v2 = -__builtin_huge_valf();
        int   i1 = 0, i2 = 0;
#pragma unroll
        for (int e = 0; e < E_DIM; ++e) {
            const float v = lds_logits[r][e];
            if (v > v1)      { v2 = v1; i2 = i1; v1 = v; i1 = e; }
            else if (v > v2) { v2 = v;  i2 = e; }
        }
        const float a2  = __expf(v2 - v1);
        const float inv = 1.0f / (1.0f + a2);
        lds_p1[r] = inv;
        lds_p2[r] = a2 * inv;
        lds_i1[r] = i1;
        lds_i2[r] = i2;
    }
    __syncthreads();

    // ---- Phase 2b: s_perm = softmax @ P_mean (256 threads, sparse: 2 terms) ----
    {
        const int r = tid >> 4, j = tid & 15;
        lds_sperm[r][j] = lds_p1[r] * lds_pm[lds_i1[r]][j]
                        + lds_p2[r] * lds_pm[lds_i2[r]][j];
    }
    __syncthreads();

    // ---- Averages: per-block partial sums -> global atomics ----
    if (tid < 32) {
        const int  j    = tid & 15;
        const bool hard = tid >= 16;
        float s = 0.0f;
#pragma unroll
        for (int r = 0; r < TILE_ROWS; ++r) {
            const float v = lds_sperm[r][j];
            s += hard ? (v >= 1e-5f ? 1.0f : 0.0f) : v;
        }
        atomicAdd(&accum[(hard ? 16 : 0) + j], s);
    }

    // ---- Phase 3: stream h tile, y[b,d] = dot(h[b,d,:], s_perm[b,:]) ----
    for (int r = 0; r < TILE_ROWS; ++r) {
        const size_t b = (size_t)(r0 + r);
        float s[E_DIM];
#pragma unroll
        for (int e = 0; e < E_DIM; ++e) s[e] = lds_sperm[r][e];

#pragma unroll
        for (int i = 0; i < D_DIM / 256; ++i) {
            const int d = i * 256 + tid;
            const float4* hp = (const float4*)(h + (b * D_DIM + d) * E_DIM);
            const float4 h0 = hp[0], h1 = hp[1], h2 = hp[2], h3 = hp[3];
            float acc;
            acc  = h0.x * s[0]  + h0.y * s[1]  + h0.z * s[2]  + h0.w * s[3];
            acc += h1.x * s[4]  + h1.y * s[5]  + h1.z * s[6]  + h1.w * s[7];
            acc += h2.x * s[8]  + h2.y * s[9]  + h2.z * s[10] + h2.w * s[11];
            acc += h3.x * s[12] + h3.y * s[13] + h3.z * s[14] + h3.w * s[15];
            y[b * D_DIM + d] = acc;
        }
    }
}

__global__ void moe_finalize_kernel(const float* __restrict__ accum,
                                    float* __restrict__ out_tail)
{
    const int t = threadIdx.x;
    if (t < 32) out_tail[t] = accum[t] * (1.0f / (float)B_TOTAL);
}

extern "C" void kernel_launch(void* const* d_in, const int* in_sizes, int n_in,
                              void* d_out, int out_size, void* d_ws, size_t ws_size,
                              hipStream_t stream) {
    (void)in_sizes; (void)n_in; (void)out_size; (void)ws_size;
    const float* h    = (const float*)d_in[0];
    const float* x    = (const float*)d_in[1];
    const float* ew   = (const float*)d_in[2];
    const float* bias = (const float*)d_in[3];
    const float* perm = (const float*)d_in[4];

    float* y     = (float*)d_out;
    float* accum = (float*)d_ws;   // 32 floats of scratch

    hipMemsetAsync(accum, 0, 32 * sizeof(float), stream);  // capture-safe

    moe_fused_kernel<<<B_TOTAL / TILE_ROWS, 256, 0, stream>>>(
        h, x, ew, bias, perm, y, accum);

    moe_finalize_kernel<<<1, 32, 0, stream>>>(
        accum, y + (size_t)B_TOTAL * D_DIM);
}